// SemiGCNConv2d_21328807592399
// MI455X (gfx1250) — compile-verified
//
#include <hip/hip_runtime.h>

typedef __attribute__((ext_vector_type(2))) float v2f;
typedef __attribute__((ext_vector_type(4))) float v4f;
typedef __attribute__((ext_vector_type(8))) float v8f;

#define BB    16
#define CIN   128
#define COUT  128
#define NN    4096
#define KNBR  16
#define LDST  130        // padded LDS row stride (floats): even (b64 align), 2-way banks max
#define NTC   64         // N columns per workgroup (4 x 16 tiles per wave)

#if defined(__has_builtin)
#if __has_builtin(__builtin_amdgcn_global_load_async_to_lds_b32)
#define HAVE_ASYNC_LDS 1
#endif
#if __has_builtin(__builtin_amdgcn_s_wait_asynccnt)
#define HAVE_WAIT_ASYNC_BUILTIN 1
#endif
#endif

typedef __attribute__((address_space(1))) int gint_t;   // global int*
typedef __attribute__((address_space(3))) int sint_t;   // LDS int*

__device__ __forceinline__ gint_t* to_global(const void* p) {
    return (gint_t*)(unsigned long long)(uintptr_t)p;
}
__device__ __forceinline__ sint_t* to_lds(void* p) {
    return (sint_t*)(unsigned int)(uintptr_t)p;   // AS3 = low 32 bits (LDS offset)
}
__device__ __forceinline__ void wait_async0() {
#if HAVE_WAIT_ASYNC_BUILTIN
    __builtin_amdgcn_s_wait_asynccnt(0);
#else
    asm volatile("s_wait_asynccnt 0x0" ::: "memory");
#endif
}

// ---------------------------------------------------------------------------
// Kernel 1: h[b][n][c] = relu( sum_k W[c][k] * x[b][k][n] )
// V_WMMA_F32_16X16X4_F32 (fp32 matrix pipe, matches reference precision).
// Each wave: M-tile = wave*16, four 16x16 accumulators (64 N columns),
// A fragment reused 4x per K-step.
// ---------------------------------------------------------------------------
__global__ __launch_bounds__(256)
void gcn_gemm_relu_kernel(const float* __restrict__ x,
                          const float* __restrict__ W,
                          float* __restrict__ h) {
    extern __shared__ float lds[];
    float* sW  = lds;                    // [COUT][LDST]
    float* sXt = lds + COUT * LDST;      // [NTC][LDST]  (transposed: [n][k])

    const int tid = threadIdx.x;
    const int b   = blockIdx.y;
    const int n0  = blockIdx.x * NTC;

    // --- stage X tile transposed: sXt[n][k] = x[b][k][n0+n] ------------------
    // 8192 elements, 256 threads -> 32 each; 64-lane-contiguous global reads.
    {
        const float* xb = x + (size_t)b * CIN * NN;
        const int t    = tid & 63;                 // n within tile
        const int krow = tid >> 6;                 // 0..3
        #pragma unroll
        for (int i = 0; i < 32; ++i) {
            int k = krow + i * 4;
#if HAVE_ASYNC_LDS
            __builtin_amdgcn_global_load_async_to_lds_b32(
                to_global(xb + (size_t)k * NN + n0 + t),
                to_lds(&sXt[t * LDST + k]), 0, 0);
#else
            sXt[t * LDST + k] = xb[(size_t)k * NN + n0 + t];
#endif
        }
    }
    // --- stage W (row-major [m][k]) into padded LDS, float4 granularity ------
    {
        const v4f* Wv = (const v4f*)W;             // 4096 float4 total
        #pragma unroll
        for (int i = 0; i < 16; ++i) {
            int e  = tid + i * 256;
            int m  = e >> 5;                       // 32 float4 per 128-float row
            int k4 = (e & 31) << 2;
            v4f w  = Wv[e];
            float* dst = &sW[m * LDST + k4];
            dst[0] = w.x; dst[1] = w.y; dst[2] = w.z; dst[3] = w.w;
        }
    }
#if HAVE_ASYNC_LDS
    wait_async0();
#endif
    __syncthreads();

    const int wave = tid >> 5;
    const int lane = tid & 31;
    const int r    = lane & 15;
    const int d    = (lane >> 4) << 1;   // 0 (lanes 0-15) or 2 (lanes 16-31)
    const int m0   = wave * 16;          // 8 waves cover all 128 output rows

    v8f acc[4] = {};

    #pragma unroll
    for (int k0 = 0; k0 < CIN; k0 += 4) {
        // A frag (16x4 f32): lane r holds M=m0+r, VGPR0/1 = K = k0+d, k0+d+1
        v2f a = *(const v2f*)&sW[(m0 + r) * LDST + k0 + d];
        #pragma unroll
        for (int nt = 0; nt < 4; ++nt) {
            // B frag (4x16 f32): lane r holds N=nt*16+r, VGPR0/1 = K+{0,1}
            v2f bfr = *(const v2f*)&sXt[(nt * 16 + r) * LDST + k0 + d];
            acc[nt] = __builtin_amdgcn_wmma_f32_16x16x4_f32(
                          false, a, false, bfr, (short)0, acc[nt], false, false);
        }
    }

    // --- relu + store: C/D layout -> lane (half hf) holds M = m0 + 8*hf + v,
    //     N = n0 + nt*16 + r.  Each lane owns 8 consecutive channels -> 2x b128.
    const int  mrow = m0 + ((lane >> 4) << 3);     // m0 or m0+8
    float*     hb   = h + (size_t)b * NN * COUT;

    #pragma unroll
    for (int nt = 0; nt < 4; ++nt) {
        float* p = &hb[(size_t)(n0 + nt * 16 + r) * COUT + mrow];
        v4f lo = { fmaxf(acc[nt][0], 0.f), fmaxf(acc[nt][1], 0.f),
                   fmaxf(acc[nt][2], 0.f), fmaxf(acc[nt][3], 0.f) };
        v4f hi = { fmaxf(acc[nt][4], 0.f), fmaxf(acc[nt][5], 0.f),
                   fmaxf(acc[nt][6], 0.f), fmaxf(acc[nt][7], 0.f) };
        *(v4f*)(p)     = lo;
        *(v4f*)(p + 4) = hi;
    }
}

// ---------------------------------------------------------------------------
// Kernel 2: out[b][c][n] = (sum_{k} h[b][idx[b][n][k]][c] + h[b][n][c]) / 17
//                          + bias[c]
// h is channel-contiguous: each neighbor is one 512B contiguous wave read
// (L2-resident: h = 33.5MB << 192MB L2). Scattered 4B stores coalesce in the
// writeback L2 since every (c,n) line is fully written.
// ---------------------------------------------------------------------------
__global__ __launch_bounds__(32)
void gcn_gather_kernel(const float* __restrict__ h,
                       const int*   __restrict__ ei0,
                       const float* __restrict__ bias,
                       float*       __restrict__ out) {
    const int n  = blockIdx.x;
    const int b  = blockIdx.y;
    const int c4 = threadIdx.x << 2;           // 32 lanes x 4 channels = 128

    const float* hb = h + (size_t)b * NN * COUT;
    const int*   e  = ei0 + ((size_t)b * NN + n) * KNBR;   // uniform -> s_loads

    v4f acc = *(const v4f*)&hb[(size_t)n * COUT + c4];     // self loop
    #pragma unroll
    for (int k = 0; k < KNBR; ++k) {
        int j = e[k];
        acc += *(const v4f*)&hb[(size_t)j * COUT + c4];
    }

    const float norm = 1.0f / (float)(KNBR + 1);
    v4f bv = *(const v4f*)&bias[c4];

    float* ob = out + (size_t)b * COUT * NN;
    ob[(size_t)(c4 + 0) * NN + n] = acc[0] * norm + bv[0];
    ob[(size_t)(c4 + 1) * NN + n] = acc[1] * norm + bv[1];
    ob[(size_t)(c4 + 2) * NN + n] = acc[2] * norm + bv[2];
    ob[(size_t)(c4 + 3) * NN + n] = acc[3] * norm + bv[3];
}

// ---------------------------------------------------------------------------
extern "C" void kernel_launch(void* const* d_in, const int* in_sizes, int n_in,
                              void* d_out, int out_size, void* d_ws, size_t ws_size,
                              hipStream_t stream) {
    const float* x    = (const float*)d_in[0];   // [B, CIN, N, 1]
    const int*   edge = (const int*)  d_in[1];   // [2, B, N, K] -> use row 0
    const float* W    = (const float*)d_in[2];   // [COUT, CIN]
    const float* bias = (const float*)d_in[3];   // [1, COUT, 1, 1]
    float*       out  = (float*)d_out;           // [B, COUT, N, 1]
    float*       h    = (float*)d_ws;            // [B][N][COUT] scratch (33.5MB)

    const size_t lds_bytes = (size_t)(COUT + NTC) * LDST * sizeof(float); // 99840B

    gcn_gemm_relu_kernel<<<dim3(NN / NTC, BB), 256, lds_bytes, stream>>>(x, W, h);
    gcn_gather_kernel  <<<dim3(NN, BB),       32, 0,          stream>>>(h, edge, bias, out);
}